// EdgeFeatureConvBlock_83236466196745
// MI455X (gfx1250) — compile-verified
//
#include <hip/hip_runtime.h>

typedef __attribute__((ext_vector_type(2))) float v2f;
typedef __attribute__((ext_vector_type(8))) float v8f;

#define PNUM 4096
#define BNUM 16
#define CNUM 32
#define K1   16
#define ENUM (PNUM * K1)
#define PPW  4                       // points per wave
#define INVF 0.9999950000374997f     // 1/sqrt(1+1e-5)

// ---------------------------------------------------------------------------
// Kernel 1: per-point precompute.
//   T[q][o]  = s0[o]*((W0a-W0b)·f)[o] + b0[o]      (o<32)   layout (B*P, 32)
//   H[q][o]  = s0[o]*(W0b·f)[o]                    (o<32)   layout (B*P, 32)
//   SC[q][o] = ssc[o]*(sc_w·f)[o] + sc_b[o]        (o<64)   layout (B*P, 64)
// ---------------------------------------------------------------------------
__global__ __launch_bounds__(256) void precompute_kernel(
    const float* __restrict__ feats, const float* __restrict__ w0,
    const float* __restrict__ g0, const float* __restrict__ b0,
    const float* __restrict__ scw, const float* __restrict__ scg,
    const float* __restrict__ scb,
    float* __restrict__ T, float* __restrict__ H, float* __restrict__ SC)
{
    int q = blockIdx.x * 256 + threadIdx.x;       // global point column
    int b = q >> 12, p = q & (PNUM - 1);
    float f[32];
    const float* fb = feats + ((size_t)b * CNUM) * PNUM + p;
#pragma unroll
    for (int c = 0; c < 32; ++c) f[c] = fb[c * PNUM];

    float* Tq = T + (size_t)q * 32;
    float* Hq = H + (size_t)q * 32;
    for (int o = 0; o < 32; ++o) {
        float a0 = 0.f, a1 = 0.f;
        const float* wr = w0 + o * 64;
#pragma unroll
        for (int c = 0; c < 32; ++c) { a0 += wr[c] * f[c]; a1 += wr[32 + c] * f[c]; }
        float s = g0[o] * INVF;
        Hq[o] = s * a1;
        Tq[o] = s * (a0 - a1) + b0[o];
    }
    float* SCq = SC + (size_t)q * 64;
    for (int o = 0; o < 64; ++o) {
        float a = 0.f;
        const float* wr = scw + o * 32;
#pragma unroll
        for (int c = 0; c < 32; ++c) a += wr[c] * f[c];
        SCq[o] = scg[o] * INVF * a + scb[o];
    }
}

// ---------------------------------------------------------------------------
// Kernel 2: fused gather + conv1 + conv2 + mean + shortcut, WMMA f32 16x16x4.
// One wave per point; 16 edges = the 16 WMMA columns.
// ---------------------------------------------------------------------------
__global__ __launch_bounds__(256) void fused_edgeconv_kernel(
    const float* __restrict__ T, const float* __restrict__ H,
    const float* __restrict__ SC, const int* __restrict__ edges,
    const float* __restrict__ w1, const float* __restrict__ g1,
    const float* __restrict__ b1, const float* __restrict__ w2,
    const float* __restrict__ g2, const float* __restrict__ b2,
    float* __restrict__ out)
{
    __shared__ __align__(16) float lds1[8][16 * 36];   // y1 C->B transpose, pitch 36
    __shared__ __align__(16) float lds2[8][64 * 20];   // conv2 out reduce,  pitch 20

    const int tid   = threadIdx.x;
    const int wave  = tid >> 5;
    const int lane  = tid & 31;       // wave32
    const int col   = lane & 15;      // N index / M row index
    const int khalf = lane >> 4;      // which K pair within a VGPR pair
    float* l1 = lds1[wave];
    float* l2 = lds2[wave];

    // --- conv1 A-tiles (scaled by s1), plus bias-as-K-extension tile -------
    v2f A1[2][8], Ab1[2];
#pragma unroll
    for (int t = 0; t < 2; ++t) {
        int m = t * 16 + col;
        float s = g1[m] * INVF;
#pragma unroll
        for (int j = 0; j < 8; ++j) {
            const float* wp = w1 + m * 32 + 4 * j + 2 * khalf;
            A1[t][j].x = wp[0] * s;      // VGPR0: K = 4j + 2*khalf
            A1[t][j].y = wp[1] * s;      // VGPR1: K = 4j + 1 + 2*khalf
        }
        Ab1[t].x = khalf ? 0.f : b1[m];  // K=32 column carries bias
        Ab1[t].y = 0.f;
    }
    v2f Bb; Bb.x = khalf ? 0.f : 1.f; Bb.y = 0.f;   // constant-1 K=32 row

    // --- conv2 A-tiles (scaled by s2) --------------------------------------
    v2f A2[4][8];
#pragma unroll
    for (int t = 0; t < 4; ++t) {
        int m = t * 16 + col;
        float s = g2[m] * INVF;
#pragma unroll
        for (int j = 0; j < 8; ++j) {
            const float* wp = w2 + m * 32 + 4 * j + 2 * khalf;
            A2[t][j].x = wp[0] * s;
            A2[t][j].y = wp[1] * s;
        }
    }
    const float bias2a = b2[lane];
    const float bias2b = b2[lane + 32];

    const int gw = blockIdx.x * 8 + wave;
    const int q0 = gw * PPW;

    for (int it = 0; it < PPW; ++it) {
        int q = q0 + it;
        int b = q >> 12, p = q & (PNUM - 1);

        // --- conv0 as gather+add+relu, directly in B-matrix layout ---------
        int tgt = edges[(size_t)b * 2 * ENUM + ENUM + p * K1 + col];
        const float* Tc = T + (size_t)q * 32 + 2 * khalf;
        const float* Hc = H + ((size_t)((b << 12) + tgt)) * 32 + 2 * khalf;
        v2f B0[8];
#pragma unroll
        for (int j = 0; j < 8; ++j) {
            v2f tp = *(const v2f*)(Tc + 4 * j);
            v2f hp = *(const v2f*)(Hc + 4 * j);
            float x = tp.x + hp.x, y = tp.y + hp.y;
            B0[j].x = x > 0.f ? x : 0.f;
            B0[j].y = y > 0.f ? y : 0.f;
        }

        // --- conv1: (32x32)·(32x16) + bias via K extension -----------------
        v8f C[2] = {};
#pragma unroll
        for (int j = 0; j < 8; ++j) {
            C[0] = __builtin_amdgcn_wmma_f32_16x16x4_f32(false, A1[0][j], false, B0[j], (short)0, C[0], false, false);
            C[1] = __builtin_amdgcn_wmma_f32_16x16x4_f32(false, A1[1][j], false, B0[j], (short)0, C[1], false, false);
        }
        C[0] = __builtin_amdgcn_wmma_f32_16x16x4_f32(false, Ab1[0], false, Bb, (short)0, C[0], false, false);
        C[1] = __builtin_amdgcn_wmma_f32_16x16x4_f32(false, Ab1[1], false, Bb, (short)0, C[1], false, false);

        // --- relu, C-layout -> LDS [col][36] (channel-major rows) ----------
#pragma unroll
        for (int t = 0; t < 2; ++t) {
            int base = t * 16 + khalf * 8;
            float* dst = l1 + col * 36 + base;
            float4 v;
            v.x = fmaxf(C[t][0], 0.f); v.y = fmaxf(C[t][1], 0.f);
            v.z = fmaxf(C[t][2], 0.f); v.w = fmaxf(C[t][3], 0.f);
            *(float4*)(dst) = v;
            v.x = fmaxf(C[t][4], 0.f); v.y = fmaxf(C[t][5], 0.f);
            v.z = fmaxf(C[t][6], 0.f); v.w = fmaxf(C[t][7], 0.f);
            *(float4*)(dst + 4) = v;
        }
        // same-wave DS ops are processed in order: read back as B-tiles
        v2f B1t[8];
#pragma unroll
        for (int j = 0; j < 8; ++j)
            B1t[j] = *(const v2f*)(l1 + col * 36 + 4 * j + 2 * khalf);

        // --- conv2: (64x32)·(32x16) ----------------------------------------
        v8f D[4] = {};
#pragma unroll
        for (int j = 0; j < 8; ++j) {
            D[0] = __builtin_amdgcn_wmma_f32_16x16x4_f32(false, A2[0][j], false, B1t[j], (short)0, D[0], false, false);
            D[1] = __builtin_amdgcn_wmma_f32_16x16x4_f32(false, A2[1][j], false, B1t[j], (short)0, D[1], false, false);
            D[2] = __builtin_amdgcn_wmma_f32_16x16x4_f32(false, A2[2][j], false, B1t[j], (short)0, D[2], false, false);
            D[3] = __builtin_amdgcn_wmma_f32_16x16x4_f32(false, A2[3][j], false, B1t[j], (short)0, D[3], false, false);
        }

        // --- dump conv2 C-tiles to LDS [o][20] for the k-mean --------------
#pragma unroll
        for (int t = 0; t < 4; ++t)
#pragma unroll
            for (int r = 0; r < 8; ++r) {
                int o = t * 16 + khalf * 8 + r;
                l2[o * 20 + col] = D[t][r];
            }

        // --- bias + relu + mean over 16 edges, fuse shortcut + final relu --
        float acc0 = 0.f, acc1 = 0.f;
        const float* r0 = l2 + lane * 20;
        const float* r1 = l2 + (lane + 32) * 20;
#pragma unroll
        for (int c4 = 0; c4 < 4; ++c4) {
            float4 va = *(const float4*)(r0 + c4 * 4);
            float4 vb = *(const float4*)(r1 + c4 * 4);
            acc0 += fmaxf(va.x + bias2a, 0.f) + fmaxf(va.y + bias2a, 0.f)
                  + fmaxf(va.z + bias2a, 0.f) + fmaxf(va.w + bias2a, 0.f);
            acc1 += fmaxf(vb.x + bias2b, 0.f) + fmaxf(vb.y + bias2b, 0.f)
                  + fmaxf(vb.z + bias2b, 0.f) + fmaxf(vb.w + bias2b, 0.f);
        }
        const float* scq = SC + (size_t)q * 64;
        float o0 = scq[lane]      + acc0 * (1.f / 16.f);
        float o1 = scq[lane + 32] + acc1 * (1.f / 16.f);
        out[((size_t)b * 64 + lane)      * PNUM + p] = fmaxf(o0, 0.f);
        out[((size_t)b * 64 + lane + 32) * PNUM + p] = fmaxf(o1, 0.f);
    }
}

// ---------------------------------------------------------------------------
extern "C" void kernel_launch(void* const* d_in, const int* in_sizes, int n_in,
                              void* d_out, int out_size, void* d_ws, size_t ws_size,
                              hipStream_t stream)
{
    const float* feats = (const float*)d_in[1];
    const int*   edges = (const int*)  d_in[2];
    const float* w0  = (const float*)d_in[3];
    const float* g0  = (const float*)d_in[4];
    const float* b0  = (const float*)d_in[5];
    const float* w1  = (const float*)d_in[6];
    const float* g1  = (const float*)d_in[7];
    const float* b1  = (const float*)d_in[8];
    const float* w2  = (const float*)d_in[9];
    const float* g2  = (const float*)d_in[10];
    const float* b2  = (const float*)d_in[11];
    const float* scw = (const float*)d_in[12];
    const float* scg = (const float*)d_in[13];
    const float* scb = (const float*)d_in[14];
    float* out = (float*)d_out;

    float* T  = (float*)d_ws;                       //  8 MB
    float* H  = T + (size_t)BNUM * PNUM * 32;       //  8 MB
    float* SC = H + (size_t)BNUM * PNUM * 32;       // 16 MB

    precompute_kernel<<<(BNUM * PNUM) / 256, 256, 0, stream>>>(
        feats, w0, g0, b0, scw, scg, scb, T, H, SC);
    fused_edgeconv_kernel<<<(BNUM * PNUM) / (8 * PPW), 256, 0, stream>>>(
        T, H, SC, edges, w1, g1, b1, w2, g2, b2, out);
}